// VoxelPooling_75290776699042
// MI455X (gfx1250) — compile-verified
//
#include <hip/hip_runtime.h>
#include <hip/hip_bf16.h>
#include <stdint.h>

// VoxelPooling: out[n,c] = mean_k src_feat[ fix(invoxel_map[n,k]) , c ]
//   fix(i) = (i == 0) ? invoxel_map[n,0] : i
// invoxel_xyz (d_in[0]) is unused by the reference -> never touched.
//
// Memory-bound (~0.4 FLOP/B): no WMMA; optimize the data path instead.
//  - async DMA of the contiguous per-block index slice into LDS (ASYNCcnt)
//  - gathers: 4 lanes x b128 = one 64B src_feat row, L2-resident (128MB < 192MB L2)
//  - non-temporal b128 stores for the output so it doesn't evict src_feat from L2

#define FUSE_K        20
#define VOX_PER_BLOCK 64
#define THREADS       256

typedef __attribute__((ext_vector_type(4))) float v4f;
typedef __attribute__((ext_vector_type(4))) int   v4i;

typedef v4i __attribute__((address_space(1))) as1_v4i;
typedef v4i __attribute__((address_space(3))) as3_v4i;

__global__ __launch_bounds__(THREADS) void voxel_pool_kernel(
    const long long* __restrict__ invoxel_map,  // [N, FUSE_K] int64
    const float*     __restrict__ src_feat,     // [M, 16] f32, rows 64B aligned
    float*           __restrict__ out,          // [N, 16] f32
    int N)
{
    __shared__ alignas(16) long long s_idx[VOX_PER_BLOCK * FUSE_K]; // 10240 B

    const int tid          = threadIdx.x;
    const int blockVoxBase = blockIdx.x * VOX_PER_BLOCK;
    const int remVox       = (N - blockVoxBase < VOX_PER_BLOCK) ? (N - blockVoxBase)
                                                                : VOX_PER_BLOCK;
    // 20 int64 per voxel = 160 B = ten 16B chunks per voxel
    const int totalChunks  = remVox * 10;

#if defined(__AMDGCN__) && __has_builtin(__builtin_amdgcn_global_load_async_to_lds_b128)
    {
        // Async global->LDS DMA of this block's index slice (tracked by ASYNCcnt).
        const char* gbase = (const char*)(invoxel_map + (size_t)blockVoxBase * FUSE_K);
        for (int i = tid; i < totalChunks; i += THREADS) {
            as1_v4i* g = (as1_v4i*)(uintptr_t)(gbase + (size_t)i * 16);
            as3_v4i* l = (as3_v4i*)(uint32_t)(uintptr_t)((char*)s_idx + (size_t)i * 16);
            __builtin_amdgcn_global_load_async_to_lds_b128(g, l, 0, 0);
        }
#if __has_builtin(__builtin_amdgcn_s_wait_asynccnt)
        __builtin_amdgcn_s_wait_asynccnt(0);
#else
        asm volatile("s_wait_asynccnt 0" ::: "memory");
#endif
    }
#else
    {
        // Fallback: plain vectorized staging copy.
        const v4i* g = (const v4i*)(invoxel_map + (size_t)blockVoxBase * FUSE_K);
        v4i*       l = (v4i*)s_idx;
        for (int i = tid; i < totalChunks; i += THREADS) l[i] = g[i];
    }
#endif
    __syncthreads();

    // 4 lanes per voxel; each lane owns one float4 chunk of the 16-channel row.
    const int vLocal = tid >> 2;
    const int chunk  = tid & 3;
    const int voxel  = blockVoxBase + vLocal;
    if (voxel < N) {
        // Index values are in [0, 2M): only the low dword of the int64 matters.
        const int* s32  = (const int*)s_idx;
        const int  base = vLocal * FUSE_K;
        const int  id0  = s32[2 * base];

        v4f acc = (v4f)0.0f;
#pragma unroll
        for (int k = 0; k < FUSE_K; ++k) {
            int id = s32[2 * (base + k)];
            id = (id == 0) ? id0 : id;
            const v4f v = *(const v4f*)(src_feat + (size_t)id * 16 + chunk * 4);
            acc += v;
        }
        const v4f r = acc * (1.0f / (float)FUSE_K);
        // Non-temporal: output (64MB) must not evict src_feat (128MB) from 192MB L2.
        __builtin_nontemporal_store(r, (v4f*)(out + (size_t)voxel * 16 + chunk * 4));
    }
}

extern "C" void kernel_launch(void* const* d_in, const int* in_sizes, int n_in,
                              void* d_out, int out_size, void* d_ws, size_t ws_size,
                              hipStream_t stream) {
    // d_in[0] = invoxel_xyz (float32, unused by reference)
    // d_in[1] = invoxel_map (int64, N*FUSE_K)
    // d_in[2] = src_feat    (float32, M*16)
    const long long* invoxel_map = (const long long*)d_in[1];
    const float*     src_feat    = (const float*)d_in[2];
    float*           out         = (float*)d_out;

    const int N      = in_sizes[1] / FUSE_K;
    const int blocks = (N + VOX_PER_BLOCK - 1) / VOX_PER_BLOCK;

    voxel_pool_kernel<<<dim3(blocks), dim3(THREADS), 0, stream>>>(
        invoxel_map, src_feat, out, N);
}